// AngularBasis_37666863186375
// MI455X (gfx1250) — compile-verified
//
#include <hip/hip_runtime.h>
#include <hip/hip_bf16.h>

// ---------------- compile-time prefactor table ----------------
constexpr double CPI = 3.141592653589793238462643383279502884;

constexpr double cfact(int n) {
    double f = 1.0;
    for (int i = 2; i <= n; ++i) f *= (double)i;
    return f;
}
constexpr double csqrt(double v) {
    double g = v > 1.0 ? v : 1.0;
    for (int i = 0; i < 120; ++i) g = 0.5 * (g + v / g);
    return g;
}
struct PfT { float v[81]; };
constexpr PfT mk_pf() {
    PfT t{};
    int idx = 0;
    for (int l = 0; l <= 8; ++l)
        for (int m = -l; m <= l; ++m) {
            int am = m < 0 ? -m : m;
            double fr = cfact(l - am) / cfact(l + am);
            double s  = csqrt(2.0) * csqrt((2.0 * l + 1.0) / (4.0 * CPI) * fr);
            t.v[idx++] = (float)(((am & 1) ? -1.0 : 1.0) * s);
        }
    return t;
}
constexpr PfT PF = mk_pf();

// 32 lanes * 81 floats per wave, 4 waves per block
#define WAVE_FLOATS 2592  // 32*81
#define BLOCK_THREADS 128

__global__ __launch_bounds__(BLOCK_THREADS)
void sh_l8_kernel(const float* __restrict__ gx, const float* __restrict__ gy,
                  const float* __restrict__ gz, const float* __restrict__ gr,
                  float* __restrict__ out, int n) {
    __shared__ float smem[4 * WAVE_FLOATS];  // 41,472 B

    const int tid  = threadIdx.x;
    const int lane = tid & 31;
    const int w    = tid >> 5;
    const int i    = blockIdx.x * BLOCK_THREADS + tid;
    const int i0   = blockIdx.x * BLOCK_THREADS + (w << 5);  // wave's first point

    const int ii = (i < n) ? i : (n - 1);
    // Streamed once, never reused -> non-temporal loads.
    const float X = __builtin_nontemporal_load(gx + ii);
    const float Y = __builtin_nontemporal_load(gy + ii);
    const float Z = __builtin_nontemporal_load(gz + ii);
    const float R = __builtin_nontemporal_load(gr + ii);

    // ---------------- per-point math (fully unrolled -> registers) ----------
    float val[81];
    {
        const float r2 = R * R;
        float q[9][9];
        q[0][0] = 1.0f;
#pragma unroll
        for (int m = 1; m < 9; ++m) q[m][m] = (float)(-(2 * m - 1)) * q[m - 1][m - 1];
#pragma unroll
        for (int m = 0; m < 8; ++m) q[m + 1][m] = (float)(2 * m + 1) * Z * q[m][m];
#pragma unroll
        for (int m = 0; m < 7; ++m) {
#pragma unroll
            for (int l = m + 2; l < 9; ++l) {
                q[l][m] = ((float)(2 * l - 1) * Z * q[l - 1][m] -
                           (float)(l + m - 1) * q[l - 2][m] * r2) *
                          (1.0f / (float)(l - m));
            }
        }
        float c[9], s[9];
        c[0] = 1.0f; s[0] = 0.0f;
#pragma unroll
        for (int m = 1; m < 9; ++m) {
            c[m] = c[m - 1] * X - s[m - 1] * Y;
            s[m] = s[m - 1] * X + c[m - 1] * Y;
        }
        const float inv_r = 1.0f / R;
        float inv_rl = 1.0f;
#pragma unroll
        for (int l = 0; l < 9; ++l) {
#pragma unroll
            for (int m = -l; m <= l; ++m) {
                const int am = m < 0 ? -m : m;
                const float phi = (m < 0) ? s[am]
                                 : (m > 0) ? c[am]
                                 : 0.70710678118654752f;
                val[l * l + (m + l)] = PF.v[l * l + (m + l)] * q[l][am] * phi * inv_rl;
            }
            inv_rl *= inv_r;
        }
    }

    // ---------------- output ------------------------------------------------
    const bool fullwave = (i0 + 32) <= n;  // wave-uniform
    if (fullwave) {
        // Stage in LDS as a byte-exact image of the 9 destination chunks.
        // Odd stride (2l+1) across lanes -> conflict-free on 64 banks.
#pragma unroll
        for (int l = 0; l < 9; ++l) {
#pragma unroll
            for (int k = 0; k < 2 * l + 1; ++k) {
                smem[w * WAVE_FLOATS + 32 * l * l + lane * (2 * l + 1) + k] =
                    val[l * l + k];
            }
        }
        // LDS writes must land before the async engine reads them.
        asm volatile("s_wait_dscnt 0x0" ::: "memory");

        const unsigned sbase = (unsigned)(unsigned long long)(&smem[0]);
#pragma unroll
        for (int l = 0; l < 9; ++l) {
            const int row = 2 * l + 1;
            const unsigned lchunk =
                sbase + (unsigned)((w * WAVE_FLOATS + 32 * l * l) * 4);
            const unsigned long long gbase =
                (unsigned long long)(out + (size_t)n * (l * l) + (size_t)i0 * row);
            // full 512B pieces: 32 lanes x 16B each
#pragma unroll
            for (int p = 0; p < row / 4; ++p) {
                unsigned long long ga = gbase + (unsigned)(p * 512 + lane * 16);
                unsigned           la = lchunk + (unsigned)(p * 512 + lane * 16);
                asm volatile("global_store_async_from_lds_b128 %0, %1, off th:TH_STORE_NT"
                             :: "v"(ga), "v"(la) : "memory");
            }
            int off = (row / 4) * 512;
            if (row & 2) {  // 256B piece
                unsigned long long ga = gbase + (unsigned)(off + lane * 8);
                unsigned           la = lchunk + (unsigned)(off + lane * 8);
                asm volatile("global_store_async_from_lds_b64 %0, %1, off th:TH_STORE_NT"
                             :: "v"(ga), "v"(la) : "memory");
                off += 256;
            }
            if (row & 1) {  // 128B piece
                unsigned long long ga = gbase + (unsigned)(off + lane * 4);
                unsigned           la = lchunk + (unsigned)(off + lane * 4);
                asm volatile("global_store_async_from_lds_b32 %0, %1, off th:TH_STORE_NT"
                             :: "v"(ga), "v"(la) : "memory");
            }
        }
        // Drain async copies before this wave retires (ENDPGM also waits-idle,
        // but be explicit so LDS reuse analysis stays obvious).
        asm volatile("s_wait_asynccnt 0x0" ::: "memory");
    } else if (i < n) {
        // Tail: bounds-checked direct non-temporal stores.
#pragma unroll
        for (int l = 0; l < 9; ++l) {
#pragma unroll
            for (int k = 0; k < 2 * l + 1; ++k) {
                __builtin_nontemporal_store(
                    val[l * l + k],
                    &out[(size_t)n * (l * l) + (size_t)i * (2 * l + 1) + k]);
            }
        }
    }
}

extern "C" void kernel_launch(void* const* d_in, const int* in_sizes, int n_in,
                              void* d_out, int out_size, void* d_ws, size_t ws_size,
                              hipStream_t stream) {
    (void)n_in; (void)out_size; (void)d_ws; (void)ws_size;
    const float* x = (const float*)d_in[0];
    const float* y = (const float*)d_in[1];
    const float* z = (const float*)d_in[2];
    const float* r = (const float*)d_in[3];
    float* out = (float*)d_out;
    const int n = in_sizes[0];
    if (n <= 0) return;
    const int grid = (n + BLOCK_THREADS - 1) / BLOCK_THREADS;
    sh_l8_kernel<<<grid, BLOCK_THREADS, 0, stream>>>(x, y, z, r, out, n);
}